// Scatt_TwoOrder_75436805587411
// MI455X (gfx1250) — compile-verified
//
#include <hip/hip_runtime.h>
#include <math.h>

// Scattering transform, 2 orders of depthwise 7x7 stride-2 convs.
// Per input channel the 16 complex filters (8 real + 8 imag) share im2col
// patches -> 16x16x4 f32 WMMA GEMM over the 49 taps (padded to 52).
// The single phi filter is accumulated on VALU from the same B registers.
// All per-chunk LDS addresses are precomputed; the xt loop is unrolled so
// the x-tile displacement folds into ds_load immediate offsets.

typedef float v2f __attribute__((ext_vector_type(2)));
typedef float v8f __attribute__((ext_vector_type(8)));

#define NB 16            // batch
#define ROWS_PER_WG 8    // output rows per workgroup (one per wave32)
#define XPW 64           // output pixels (x) per workgroup
#define NXT (XPW / 16)   // x-tiles of 16 per wave
#define IN_ROWS 21       // 2*(ROWS_PER_WG-1) + 7
#define IN_COLS 133      // 2*(XPW-1) + 7
#define IN_PITCH 136
#define NCHUNK 13        // ceil(49/4) K-chunks of 4 taps

template<int ORDER, int CHI, int H_IN, int W_IN, int H_OUT, int W_OUT>
__global__ __launch_bounds__(256)
void scatter_order(const float* __restrict__ in,     // [NB][CHI][H_IN][W_IN]
                   const float* __restrict__ phi_w,  // [3][49]
                   const float* __restrict__ psi_r,  // [24][49]
                   const float* __restrict__ psi_i,  // [24][49]
                   float* __restrict__ out)
{
    __shared__ float sIn[IN_ROWS * IN_PITCH];   // input halo tile
    __shared__ float sWa[16 * 52];              // 16 filters x 52 taps (zero padded)
    __shared__ float sWp[52];                   // phi filter, zero padded

    const int tid = threadIdx.x;
    const int nXB = W_OUT / XPW;
    const int nYB = H_OUT / ROWS_PER_WG;
    int bid = blockIdx.x;
    const int xb = bid % nXB; bid /= nXB;
    const int yb = bid % nYB; bid /= nYB;
    const int cc = bid % CHI; bid /= CHI;
    const int b  = bid;

    // order-2 weights are tiled: base filter set = cc % 3
    const int bc = (ORDER == 1) ? cc : (cc % 3);

    // ---- weights -> LDS (filters: 0..7 = real, 8..15 = imag) ----
    for (int idx = tid; idx < 16 * 52; idx += 256) {
        const int f = idx / 52, t = idx % 52;
        float v = 0.f;
        if (t < 49)
            v = (f < 8) ? psi_r[(bc * 8 + f) * 49 + t]
                        : psi_i[(bc * 8 + (f - 8)) * 49 + t];
        sWa[idx] = v;
    }
    for (int idx = tid; idx < 52; idx += 256)
        sWp[idx] = (idx < 49) ? phi_w[bc * 49 + idx] : 0.f;

    // ---- input halo -> LDS with zero padding ----
    const int y0 = yb * ROWS_PER_WG;
    const int x0 = xb * XPW;
    const long long plane_in = (long long)H_IN * W_IN;
    const float* inC = in + ((long long)b * CHI + cc) * plane_in;
    const int gr0 = 2 * y0 - 3, gc0 = 2 * x0 - 3;
    for (int idx = tid; idx < IN_ROWS * IN_PITCH; idx += 256) {
        const int r = idx / IN_PITCH, cl = idx % IN_PITCH;
        const int gr = gr0 + r, gc = gc0 + cl;
        float v = 0.f;
        if (cl < IN_COLS && gr >= 0 && gr < H_IN && gc >= 0 && gc < W_IN)
            v = inC[(long long)gr * W_IN + gc];
        sIn[idx] = v;
    }
    __syncthreads();

    const int lane = tid & 31;   // wave32
    const int wave = tid >> 5;   // 0..7 -> output row within block
    const int n    = lane & 15;  // M (filter) index for A, N (pixel) index for B
    const int sub  = lane >> 4;  // 0: K sub-taps {0,1}; 1: K sub-taps {2,3}

    // Preload A-matrix (weights), phi weights, and per-chunk LDS addresses.
    // 32-bit A 16x4 layout: lanes 0-15 M=0..15 VGPR0=K0/VGPR1=K1,
    //                       lanes 16-31 M=0..15 VGPR0=K2/VGPR1=K3.
    const int rowBase = 2 * wave;          // LDS input row for ky=0
    float a0[NCHUNK], a1[NCHUNK], p0[NCHUNK], p1[NCHUNK];
    int addr0[NCHUNK], addr1[NCHUNK];      // LDS float index for xt=0, kx folded in
#pragma unroll
    for (int ch = 0; ch < NCHUNK; ++ch) {
        const int t = 4 * ch + 2 * sub;
        a0[ch] = sWa[n * 52 + t];
        a1[ch] = sWa[n * 52 + t + 1];
        p0[ch] = sWp[t];
        p1[ch] = sWp[t + 1];
        int t0 = t, t1 = t + 1;
        if (t0 > 48) t0 = 48;   // padded taps: A/p are zero there anyway
        if (t1 > 48) t1 = 48;
        const int ky0 = t0 / 7, kx0 = t0 - 7 * ky0;
        const int ky1 = t1 / 7, kx1 = t1 - 7 * ky1;
        addr0[ch] = (rowBase + ky0) * IN_PITCH + 2 * n + kx0;
        addr1[ch] = (rowBase + ky1) * IN_PITCH + 2 * n + kx1;
    }

    const int y = y0 + wave;
    const long long plane_out = (long long)H_OUT * W_OUT;

#pragma unroll
    for (int xt = 0; xt < NXT; ++xt) {
        v8f acc = {0.f, 0.f, 0.f, 0.f, 0.f, 0.f, 0.f, 0.f};
        float phiAcc = 0.f;
        const int xoff = xt * 32;            // +32 floats per x-tile (imm offset)
#pragma unroll
        for (int ch = 0; ch < NCHUNK; ++ch) {
            const float b0 = sIn[addr0[ch] + xoff];
            const float b1 = sIn[addr1[ch] + xoff];
            v2f A  = { a0[ch], a1[ch] };
            v2f Bv = { b0, b1 };
            // D[m][n] += A(16x4) x B(4x16): m=filter, n=pixel, K=4 taps
            acc = __builtin_amdgcn_wmma_f32_16x16x4_f32(
                      false, A, false, Bv, (short)0, acc, false, false);
            // phi filter on VALU, reusing the same patch values
            phiAcc = fmaf(p0[ch], b0, phiAcc);
            phiAcc = fmaf(p1[ch], b1, phiAcc);
        }
        // combine lane halves: lanes<16 hold taps{0,1} partials / r_j,
        // lanes>=16 hold taps{2,3} partials / i_j
        const float phiOut = phiAcc + __shfl_xor(phiAcc, 16, 32);
        float u[8];
#pragma unroll
        for (int j = 0; j < 8; ++j) {
            const float rv = acc[j];
            const float ov = __shfl_xor(rv, 16, 32);
            u[j] = sqrtf(fmaf(rv, rv, ov * ov));
        }

        const int x = x0 + xt * 16 + n;
        const long long pix = (long long)y * W_OUT + x;
        long long basePhi, basePsi;
        if (ORDER == 1) {
            // s1 channel layout: c*9 + {0:phi, 1+j:psi_j}
            basePhi = ((long long)b * 27 + cc * 9) * plane_out + pix;
            basePsi = basePhi + plane_out;
        } else {
            // s2 channel layout: co*81 + {k:phi, 9 + k*8 + j:psi_j}, k = cc%9
            const int co = cc / 9, k = cc - co * 9;
            basePhi = ((long long)b * 243 + co * 81 + k) * plane_out + pix;
            basePsi = ((long long)b * 243 + co * 81 + 9 + k * 8) * plane_out + pix;
        }
        if (sub == 0) {
            out[basePhi] = phiOut;
#pragma unroll
            for (int j = 0; j < 4; ++j)
                out[basePsi + (long long)j * plane_out] = u[j];
        } else {
#pragma unroll
            for (int j = 4; j < 8; ++j)
                out[basePsi + (long long)j * plane_out] = u[j];
        }
    }
}

extern "C" void kernel_launch(void* const* d_in, const int* in_sizes, int n_in,
                              void* d_out, int out_size, void* d_ws, size_t ws_size,
                              hipStream_t stream) {
    const float* x   = (const float*)d_in[0];   // [16,3,512,512]
    const float* phw = (const float*)d_in[1];   // [3,1,7,7]
    const float* prw = (const float*)d_in[2];   // [24,1,7,7]
    const float* piw = (const float*)d_in[3];   // [24,1,7,7]
    float* s2 = (float*)d_out;                  // [16,243,128,128]
    float* s1 = (float*)d_ws;                   // [16,27,256,256] scratch (113 MB)

    dim3 blk(256, 1, 1);

    // Order 1: x -> s1
    dim3 g1(NB * 3 * (256 / ROWS_PER_WG) * (256 / XPW), 1, 1);   // 6144 blocks
    scatter_order<1, 3, 512, 512, 256, 256><<<g1, blk, 0, stream>>>(
        x, phw, prw, piw, s1);

    // Order 2: s1 -> s2
    dim3 g2(NB * 27 * (128 / ROWS_PER_WG) * (128 / XPW), 1, 1);  // 13824 blocks
    scatter_order<2, 27, 256, 256, 128, 128><<<g2, blk, 0, stream>>>(
        s1, phw, prw, piw, s2);
}